// ConvCRF3d_26233660244509
// MI455X (gfx1250) — compile-verified
//
#include <hip/hip_runtime.h>

// ---------------------------------------------------------------------------
// ConvCRF3d mean-field, 96^3 volume, 4 classes, 27-tap kernels, 3 iterations.
// Exploits: k_pos spatially constant; combined weights iteration-invariant
// (recomputed per launch to avoid a 96MB intermediate). Message aggregation
// done with V_WMMA_F32_16X16X4_F32 (full fp32 precision). Tiles staged into
// LDS with GLOBAL_LOAD_ASYNC_TO_LDS_B32 when the builtin is available.
// ---------------------------------------------------------------------------

#define NW   96
#define NWH  (96 * 96)          // 9216
#define NVOL (96 * 96 * 96)     // 884736

typedef __attribute__((ext_vector_type(2))) float v2f;
typedef __attribute__((ext_vector_type(8))) float v8f;

#if defined(__gfx1250__) && __has_builtin(__builtin_amdgcn_global_load_async_to_lds_b32)
#define USE_ASYNC_LDS 1
#else
#define USE_ASYNC_LDS 0
#endif

__device__ __forceinline__ void copy_g2lds(const float* g, float* l) {
#if USE_ASYNC_LDS
  float* gm = const_cast<float*>(g);
  __builtin_amdgcn_global_load_async_to_lds_b32(
      (__attribute__((address_space(1))) int*)(void*)gm,
      (__attribute__((address_space(3))) int*)(void*)l,
      0, 0);
#else
  *l = *g;
#endif
}

__device__ __forceinline__ void wait_async_then_barrier() {
#if USE_ASYNC_LDS
#if __has_builtin(__builtin_amdgcn_s_wait_asynccnt)
  __builtin_amdgcn_s_wait_asynccnt(0);
#else
  asm volatile("s_wait_asynccnt 0" ::: "memory");
#endif
#endif
  __syncthreads();
}

// Tile geometry: core 16(x) x 8(y) x 1(z); halo'd source region 18 x 10 x 3.
#define TSX 18
#define TSY 10
#define TSZ 3
#define TSLOT (TSX * TSY * TSZ)  // 540

__global__ __launch_bounds__(128) void crf_iter_kernel(
    const float* __restrict__ img,
    const float* __restrict__ spacing,    // 3
    const float* __restrict__ ptheta,     // 3 (z,y,x)
    const float* __restrict__ ctheta,     // 1
    const float* __restrict__ posw,       // 1
    const float* __restrict__ colw,       // 1
    const float* __restrict__ unary,      // 4*VOL
    const float* __restrict__ pred,       // 4*VOL (previous prediction)
    float* __restrict__ dst)              // 4*VOL
{
  __shared__ float sImg[TSLOT];
  __shared__ float sPred[4 * TSLOT];
  __shared__ float sComb[128 * 27];

  const int t   = threadIdx.x;
  const int gx0 = blockIdx.x * 16;
  const int gy0 = blockIdx.y * 8;
  const int gz  = blockIdx.z;

  // ---------------- Phase 0: stage halo'd tiles (img + 4 pred ch) to LDS ---
  for (int i = t; i < 5 * TSLOT; i += 128) {
    const int ch   = i / TSLOT;       // 0 = img, 1..4 = pred ch-1
    const int slot = i - ch * TSLOT;
    const int xl = slot % TSX;
    const int yl = (slot / TSX) % TSY;
    const int zl = slot / (TSX * TSY);
    const int gx = gx0 - 1 + xl;
    const int gy = gy0 - 1 + yl;
    const int gzz = gz - 1 + zl;
    float* lp = (ch == 0) ? &sImg[slot] : &sPred[(ch - 1) * TSLOT + slot];
    const bool ok = ((unsigned)gx < 96u) && ((unsigned)gy < 96u) && ((unsigned)gzz < 96u);
    if (ok) {
      const long gidx = (long)gzz * NWH + gy * NW + gx;
      const float* gp = (ch == 0) ? (img + gidx) : (pred + (long)(ch - 1) * NVOL + gidx);
      copy_g2lds(gp, lp);
    } else {
      *lp = 0.0f;
    }
  }
  wait_async_then_barrier();

  // ---------------- Phase A: per-voxel combined 27-tap weights -> LDS ------
  const float sz = spacing[0], sy = spacing[1], sx = spacing[2];
  const float tz = ptheta[0], ty = ptheta[1], tx = ptheta[2];
  const float tc = ctheta[0];
  const float pw = posw[0], cw = colw[0];
  // k_pos is spatially constant: per-axis factors exp(-0.5*theta*(s/8)^2)
  const float ez = __expf(-0.5f * tz * (sz * 0.125f) * (sz * 0.125f));
  const float ey = __expf(-0.5f * ty * (sy * 0.125f) * (sy * 0.125f));
  const float ex = __expf(-0.5f * tx * (sx * 0.125f) * (sx * 0.125f));
  const float nTc = -2.0f * tc;  // feats_col = img/0.5 -> exp(-2*theta*dI^2)

  {
    const int mx = t & 15;
    const int my = t >> 4;                 // 0..7
    const int gx = gx0 + mx;
    const int gy = gy0 + my;
    const int base27 = t * 27;
    const float Ic = sImg[1 * (TSX * TSY) + (my + 1) * TSX + (mx + 1)];

    float sumc = 0.0f, sump = 0.0f;
#pragma unroll
    for (int dz = 0; dz < 3; ++dz) {
#pragma unroll
      for (int dy = 0; dy < 3; ++dy) {
#pragma unroll
        for (int dx = 0; dx < 3; ++dx) {
          const int didx = dz * 9 + dy * 3 + dx;
          const float In = sImg[dz * (TSX * TSY) + (my + dy) * TSX + (mx + dx)];
          const float diff = Ic - In;
          const float w = __expf(nTc * diff * diff);
          const float p = (dz == 1 ? 1.0f : ez) * (dy == 1 ? 1.0f : ey) *
                          (dx == 1 ? 1.0f : ex);
          const bool v = ((unsigned)(gx + dx - 1) < 96u) &&
                         ((unsigned)(gy + dy - 1) < 96u) &&
                         ((unsigned)(gz + dz - 1) < 96u);
          sComb[base27 + didx] = w;
          sumc += v ? w : 0.0f;
          sump += v ? p : 0.0f;
        }
      }
    }
    const float ra = pw / (sump + 1e-20f);
    const float rb = cw / (sumc + 1e-20f);
#pragma unroll
    for (int dz = 0; dz < 3; ++dz) {
#pragma unroll
      for (int dy = 0; dy < 3; ++dy) {
#pragma unroll
        for (int dx = 0; dx < 3; ++dx) {
          const int didx = dz * 9 + dy * 3 + dx;
          const float p = (dz == 1 ? 1.0f : ez) * (dy == 1 ? 1.0f : ey) *
                          (dx == 1 ? 1.0f : ex);
          const bool v = ((unsigned)(gx + dx - 1) < 96u) &&
                         ((unsigned)(gy + dy - 1) < 96u) &&
                         ((unsigned)(gz + dz - 1) < 96u);
          const float w = sComb[base27 + didx];
          sComb[base27 + didx] = v ? (ra * p + rb * w) : 0.0f;
        }
      }
    }
  }
  __syncthreads();

  // ---------------- Phase B: WMMA f32 16x16x4 message aggregation ----------
  // Wave wv handles rows my = 2*wv, 2*wv+1. Lane layout (A 16x4, f32):
  //   lane l: row m = l&15, half = l>>4; V0 = A[m, 2*half], V1 = A[m, 2*half+1]
  // B 4x16: lane l: col n = l&15; V0 = B[2*half, n], V1 = B[2*half+1, n]
  // D 16x16 f32: VGPR r: lanes 0-15 -> M=r, lanes 16-31 -> M=8+r; N=lane&15
  const int lane = t & 31;
  const int wv   = t >> 5;        // 0..3
  const int m    = lane & 15;     // output-voxel row (A) / class column (B,D)
  const int half = lane >> 4;
  const int ccls = m;             // class index for B/D columns (valid if <4)

  for (int rr = 0; rr < 2; ++rr) {
    const int myr = wv * 2 + rr;         // core row 0..7
    const int vox27 = (myr * 16 + m) * 27;
    v8f acc = {0.f, 0.f, 0.f, 0.f, 0.f, 0.f, 0.f, 0.f};

#pragma unroll
    for (int dz = 0; dz < 3; ++dz) {
#pragma unroll
      for (int dy = 0; dy < 3; ++dy) {
        const int prow  = dz * (TSX * TSY) + (myr + dy) * TSX;  // + s gives tile idx
        const int dbase = dz * 9 + dy * 3;
#pragma unroll
        for (int ch4 = 0; ch4 < 5; ++ch4) {
          const int s0 = ch4 * 4 + 2 * half;   // source index 0..19 (valid <=17)
          const int s1 = s0 + 1;
          const int sc0 = (s0 > 17) ? 17 : s0;
          const int sc1 = (s1 > 17) ? 17 : s1;

          // B: pred[class, source]
          const float pv0 = sPred[(ccls & 3) * TSLOT + prow + sc0];
          const float pv1 = sPred[(ccls & 3) * TSLOT + prow + sc1];
          const float b0 = (ccls < 4 && s0 <= 17) ? pv0 : 0.0f;
          const float b1 = (ccls < 4 && s1 <= 17) ? pv1 : 0.0f;

          // A: comb weight (out voxel m, offset dx = s-1-m), banded |dx|<=1
          const int dx0 = s0 - 1 - m;
          const int dx1 = s1 - 1 - m;
          const int dc0 = dx0 < -1 ? -1 : (dx0 > 1 ? 1 : dx0);
          const int dc1 = dx1 < -1 ? -1 : (dx1 > 1 ? 1 : dx1);
          const float aw0 = sComb[vox27 + dbase + (dc0 + 1)];
          const float aw1 = sComb[vox27 + dbase + (dc1 + 1)];
          const float a0 = (dx0 >= -1 && dx0 <= 1 && s0 <= 17) ? aw0 : 0.0f;
          const float a1 = (dx1 >= -1 && dx1 <= 1 && s1 <= 17) ? aw1 : 0.0f;

          v2f A = {a0, a1};
          v2f B = {b0, b1};
          acc = __builtin_amdgcn_wmma_f32_16x16x4_f32(
              false, A, false, B, (short)0, acc, false, false);
        }
      }
    }

    // ---------------- momentum + writeback (lanes with class < 4) ----------
    if (ccls < 4) {
      const int gy = gy0 + myr;
#pragma unroll
      for (int r = 0; r < 8; ++r) {
        const int mprime = r + 8 * half;
        const long idx = (long)ccls * NVOL + (long)gz * NWH + gy * NW + (gx0 + mprime);
        dst[idx] = 0.8f * unary[idx] + 0.2f * acc[r];
      }
    }
  }
}

extern "C" void kernel_launch(void* const* d_in, const int* in_sizes, int n_in,
                              void* d_out, int out_size, void* d_ws, size_t ws_size,
                              hipStream_t stream) {
  (void)in_sizes; (void)n_in; (void)out_size; (void)ws_size;
  const float* image     = (const float*)d_in[0];
  const float* unary     = (const float*)d_in[1];
  const float* spacing   = (const float*)d_in[2];
  const float* pos_theta = (const float*)d_in[3];
  const float* col_theta = (const float*)d_in[4];
  const float* pos_w     = (const float*)d_in[5];
  const float* col_w     = (const float*)d_in[6];
  float* out = (float*)d_out;
  float* tmp = (float*)d_ws;   // 4*96^3 floats of ping-pong scratch

  dim3 grid(96 / 16, 96 / 8, 96);
  dim3 block(128);

  // iter 1: pred = unary -> out
  crf_iter_kernel<<<grid, block, 0, stream>>>(image, spacing, pos_theta, col_theta,
                                              pos_w, col_w, unary, unary, out);
  // iter 2: out -> tmp
  crf_iter_kernel<<<grid, block, 0, stream>>>(image, spacing, pos_theta, col_theta,
                                              pos_w, col_w, unary, out, tmp);
  // iter 3: tmp -> out (final result in d_out)
  crf_iter_kernel<<<grid, block, 0, stream>>>(image, spacing, pos_theta, col_theta,
                                              pos_w, col_w, unary, tmp, out);
}